// SMPL_49409303773846
// MI455X (gfx1250) — compile-verified
//
#include <hip/hip_runtime.h>

#define V_CNT  6890
#define NJ     24
#define BATCH  512
#define M3     (V_CNT * 3)        // 20670 rows of the fused GEMM
#define KP     207                // posedirs K
#define KSTEPS 55                 // 220 combined K / 4
#define KMAIN  51                 // k-steps 0..50 are pure posedirs (k <= 206)
#define MTILES ((M3 + 15) / 16)   // 1292

typedef float v2f __attribute__((ext_vector_type(2)));
typedef float v8f __attribute__((ext_vector_type(8)));

// parent of joint j (j=0 unused)
__constant__ int c_par[NJ] = {0, 0, 0, 0, 1, 2, 3, 4, 5, 6, 7, 8,
                              9, 9, 9, 12, 13, 14, 16, 17, 18, 19, 20, 21};

// B stored in WMMA B-fragment order: element for (k, b) lives at
// Bfrag[((nt*55 + k/4)*32 + (k%4/2)*16 + b%16)*2 + k%2],  nt = b/16
__device__ __forceinline__ void bfrag_store(float* __restrict__ Bfrag,
                                            int k, int b, float val) {
  const int nt = b >> 4, mr = b & 15;
  const int kstep = k >> 2, rem = k & 3;
  const size_t idx =
      ((size_t)(nt * KSTEPS + kstep) * 32 + ((rem >> 1) * 16 + mr)) * 2 + (rem & 1);
  Bfrag[idx] = val;
}

// ---------------------------------------------------------------------------
// Kernel 1: batch-independent precompute
//   JS[j][k][d] = sum_v Jreg[j,v] * shapedirs[(v*3+k), d]
//   Jt[j][k]    = sum_v Jreg[j,v] * v_template[v, k]
// ---------------------------------------------------------------------------
__global__ __launch_bounds__(256) void smpl_prep_joints(
    const float* __restrict__ Jreg, const float* __restrict__ shapedirs,
    const float* __restrict__ vtemp, float* __restrict__ JS, float* __restrict__ Jt) {
  const int j = blockIdx.x;
  const int t = threadIdx.x;
  float acc[33];
#pragma unroll
  for (int i = 0; i < 33; ++i) acc[i] = 0.f;
  for (int v = t; v < V_CNT; v += 256) {
    const float w = Jreg[j * V_CNT + v];
#pragma unroll
    for (int k = 0; k < 3; ++k) {
      acc[30 + k] += w * vtemp[v * 3 + k];
      const float* sd = shapedirs + (size_t)(v * 3 + k) * 10;
#pragma unroll
      for (int d = 0; d < 10; ++d) acc[k * 10 + d] += w * sd[d];
    }
  }
  __shared__ float s[256 * 33];
#pragma unroll
  for (int i = 0; i < 33; ++i) s[t * 33 + i] = acc[i];
  __syncthreads();
  if (t < 33) {
    float sum = 0.f;
    for (int u = 0; u < 256; ++u) sum += s[u * 33 + t];
    if (t < 30) JS[j * 30 + t] = sum;
    else        Jt[j * 3 + (t - 30)] = sum;
  }
}

// ---------------------------------------------------------------------------
// Kernel 2: one wave32 per batch element.
//   rodrigues -> R; scatter B (lrotmin|beta|1|0) into fragment order;
//   Jnts = Jt + JS*beta; local transforms; serial FK; G_rel -> ws.
// ---------------------------------------------------------------------------
__global__ __launch_bounds__(32) void smpl_batch_small(
    const float* __restrict__ pose, const float* __restrict__ beta,
    const float* __restrict__ JS, const float* __restrict__ Jt,
    float* __restrict__ Bfrag, float* __restrict__ Grel) {
  const int b = blockIdx.x;
  const int j = threadIdx.x;

  __shared__ float sJ[NJ * 3];
  __shared__ float sGloc[NJ * 12];
  __shared__ float sG[NJ * 12];

  float R[9];
  if (j < NJ) {
    const float rx = pose[b * 72 + j * 3 + 0];
    const float ry = pose[b * 72 + j * 3 + 1];
    const float rz = pose[b * 72 + j * 3 + 2];
    const float th = sqrtf(rx * rx + ry * ry + rz * rz + 1e-8f);
    const float ux = rx / th, uy = ry / th, uz = rz / th;
    const float c = cosf(th), s = sinf(th), ic = 1.f - c;
    R[0] = c + ic * ux * ux; R[1] = ic * ux * uy - s * uz; R[2] = ic * ux * uz + s * uy;
    R[3] = ic * uy * ux + s * uz; R[4] = c + ic * uy * uy; R[5] = ic * uy * uz - s * ux;
    R[6] = ic * uz * ux - s * uy; R[7] = ic * uz * uy + s * ux; R[8] = c + ic * uz * uz;

    if (j >= 1) {  // lrotmin rows (joints 1..23)
      const int p0 = (j - 1) * 9;
#pragma unroll
      for (int e = 0; e < 9; ++e) {
        const float diag = (e == 0 || e == 4 || e == 8) ? 1.f : 0.f;
        bfrag_store(Bfrag, p0 + e, b, R[e] - diag);
      }
    }
#pragma unroll
    for (int k = 0; k < 3; ++k) {  // joints via precomputed affine map
      float v = Jt[j * 3 + k];
      const float* js = JS + (j * 3 + k) * 10;
#pragma unroll
      for (int d = 0; d < 10; ++d) v += js[d] * beta[b * 10 + d];
      sJ[j * 3 + k] = v;
    }
  }
  if (j < 10) bfrag_store(Bfrag, KP + j, b, beta[b * 10 + j]);
  if (j == 0) {
    bfrag_store(Bfrag, 217, b, 1.f);
    bfrag_store(Bfrag, 218, b, 0.f);
    bfrag_store(Bfrag, 219, b, 0.f);
  }
  __syncthreads();

  if (j < NJ) {
    const int p = c_par[j];
#pragma unroll
    for (int m = 0; m < 3; ++m) {
#pragma unroll
      for (int n = 0; n < 3; ++n) sGloc[j * 12 + m * 4 + n] = R[m * 3 + n];
      const float jm = sJ[j * 3 + m];
      sGloc[j * 12 + m * 4 + 3] = (j == 0) ? jm : (jm - sJ[p * 3 + m]);
    }
  }
  __syncthreads();

  if (j == 0) {  // serial forward kinematics
#pragma unroll
    for (int e = 0; e < 12; ++e) sG[e] = sGloc[e];
    for (int i = 1; i < NJ; ++i) {
      const int p = c_par[i];
      for (int m = 0; m < 3; ++m) {
        for (int n = 0; n < 4; ++n) {
          float v = (n == 3) ? sG[p * 12 + m * 4 + 3] : 0.f;
          for (int k = 0; k < 3; ++k) v += sG[p * 12 + m * 4 + k] * sGloc[i * 12 + k * 4 + n];
          sG[i * 12 + m * 4 + n] = v;
        }
      }
    }
  }
  __syncthreads();

  if (j < NJ) {
    const float jx = sJ[j * 3 + 0], jy = sJ[j * 3 + 1], jz = sJ[j * 3 + 2];
#pragma unroll
    for (int m = 0; m < 3; ++m) {
      const float g0 = sG[j * 12 + m * 4 + 0];
      const float g1 = sG[j * 12 + m * 4 + 1];
      const float g2 = sG[j * 12 + m * 4 + 2];
      const float g3 = sG[j * 12 + m * 4 + 3];
      float* o = Grel + (size_t)b * (NJ * 12) + j * 12 + m * 4;
      o[0] = g0; o[1] = g1; o[2] = g2;
      o[3] = g3 - (g0 * jx + g1 * jy + g2 * jz);
    }
  }
}

// ---------------------------------------------------------------------------
// Kernel 3: pack the mixed K-tail (k = 204..219) of the combined A matrix:
//   Atail[v3][c] , c=0..15 -> k=204+c : posedirs | shapedirs | v_template | 0
// ---------------------------------------------------------------------------
__global__ __launch_bounds__(256) void smpl_pack_atail(
    const float* __restrict__ posedirs, const float* __restrict__ shapedirs,
    const float* __restrict__ vtemp, float* __restrict__ Atail) {
  const int idx = blockIdx.x * 256 + threadIdx.x;  // grid is floor(M3*16/256) blocks: last 224 of M3*16 unwritten
  const int v3 = idx >> 4;
  const int k = 204 + (idx & 15);
  float val;
  if (k < KP)            val = posedirs[(size_t)v3 * KP + k];
  else if (k < KP + 10)  val = shapedirs[(size_t)v3 * 10 + (k - KP)];
  else if (k == KP + 10) val = vtemp[v3];
  else                   val = 0.f;
  Atail[idx] = val;
}

// ---------------------------------------------------------------------------
// Kernel 4: fused WMMA GEMM (exact f32, V_WMMA_F32_16X16X4_F32)
//   v_posedT[v3][b] = sum_k A[v3][k] * B[k][b]
//   Branch-free inner loops: main region straight from posedirs (row clamp,
//   no EXEC games), tail from packed Atail; B fragments are aligned b64.
// ---------------------------------------------------------------------------
__global__ __launch_bounds__(256) void smpl_gemm(
    const float* __restrict__ posedirs, const float* __restrict__ Atail,
    const float* __restrict__ Bfrag, float* __restrict__ vpT) {
  const int lane = threadIdx.x & 31;
  const int wave = threadIdx.x >> 5;
  const int mt = blockIdx.x * 8 + wave;
  if (mt >= MTILES) return;           // wave-uniform: EXEC all-ones at every WMMA
  const int nt = blockIdx.y;
  const int half = lane >> 4;         // lanes 16..31 hold K+2,K+3 of each step
  const int mr = lane & 15;
  const int vt16 = mt * 16;
  const int bt = nt * 16;
  int v3 = vt16 + mr;
  if (v3 > M3 - 1) v3 = M3 - 1;       // clamp: pad rows compute garbage, never stored

  const float* arow  = posedirs + (size_t)v3 * KP + half * 2;
  const float* brow  = Bfrag + ((size_t)nt * KSTEPS * 32 + lane) * 2;
  const float* atail = Atail + ((size_t)v3 << 4) + half * 2;

  v8f acc = {};
#pragma unroll 3
  for (int ks = 0; ks < KMAIN; ++ks) {
    v2f a, bb;
    a.x = arow[ks * 4 + 0];
    a.y = arow[ks * 4 + 1];
    bb = *(const v2f*)(brow + (size_t)ks * 64);
    acc = __builtin_amdgcn_wmma_f32_16x16x4_f32(false, a, false, bb,
                                                (short)0, acc, false, false);
  }
#pragma unroll
  for (int ks = KMAIN; ks < KSTEPS; ++ks) {
    v2f a  = *(const v2f*)(atail + (ks - KMAIN) * 4);
    v2f bb = *(const v2f*)(brow + (size_t)ks * 64);
    acc = __builtin_amdgcn_wmma_f32_16x16x4_f32(false, a, false, bb,
                                                (short)0, acc, false, false);
  }
  // C layout: VGPR r, lanes 0-15 -> M=r, lanes 16-31 -> M=r+8; N = lane&15
#pragma unroll
  for (int r = 0; r < 8; ++r) {
    const int row = vt16 + r + half * 8;
    if (row < M3) vpT[(size_t)row * BATCH + bt + mr] = acc[r];
  }
}

// ---------------------------------------------------------------------------
// Kernel 5: LBS skinning. Block = (v-chunk, b). G_rel[b] staged in LDS.
// ---------------------------------------------------------------------------
__global__ __launch_bounds__(256) void smpl_skin(
    const float* __restrict__ W, const float* __restrict__ Grel,
    const float* __restrict__ vpT, float* __restrict__ out) {
  const int b = blockIdx.y;
  const int t = threadIdx.x;
  __shared__ float sG[NJ * 12];
  for (int i = t; i < NJ * 12; i += 256) sG[i] = Grel[(size_t)b * (NJ * 12) + i];
  __syncthreads();
  const int v = blockIdx.x * 256 + t;
  if (v >= V_CNT) return;

  float T[12];
#pragma unroll
  for (int e = 0; e < 12; ++e) T[e] = 0.f;
  const float* wv = W + (size_t)v * NJ;
#pragma unroll 4
  for (int jj = 0; jj < NJ; ++jj) {
    const float w = wv[jj];
#pragma unroll
    for (int e = 0; e < 12; ++e) T[e] += w * sG[jj * 12 + e];
  }
  const float px = vpT[(size_t)(3 * v + 0) * BATCH + b];
  const float py = vpT[(size_t)(3 * v + 1) * BATCH + b];
  const float pz = vpT[(size_t)(3 * v + 2) * BATCH + b];
#pragma unroll
  for (int m = 0; m < 3; ++m)
    out[(size_t)b * M3 + 3 * v + m] =
        T[m * 4 + 0] * px + T[m * 4 + 1] * py + T[m * 4 + 2] * pz + T[m * 4 + 3];
}

// ---------------------------------------------------------------------------
extern "C" void kernel_launch(void* const* d_in, const int* in_sizes, int n_in,
                              void* d_out, int out_size, void* d_ws, size_t ws_size,
                              hipStream_t stream) {
  const float* pose      = (const float*)d_in[0];
  const float* beta      = (const float*)d_in[1];
  const float* Jreg      = (const float*)d_in[2];
  const float* W         = (const float*)d_in[3];
  const float* posedirs  = (const float*)d_in[4];
  const float* vtemp     = (const float*)d_in[5];
  const float* shapedirs = (const float*)d_in[6];

  float* ws    = (float*)d_ws;
  float* JS    = ws;                    // 720
  float* Jt    = ws + 720;              // 72
  float* Bfrag = ws + 792;              // 32*55*32*2  = 112640
  float* Grel  = ws + 113432;           // 512*288     = 147456
  float* Atail = ws + 260888;           // 20670*16    = 330720
  float* vpT   = ws + 591608;           // 20670*512   = 10583040  (~44.7 MB total)

  smpl_prep_joints<<<NJ, 256, 0, stream>>>(Jreg, shapedirs, vtemp, JS, Jt);
  smpl_batch_small<<<BATCH, 32, 0, stream>>>(pose, beta, JS, Jt, Bfrag, Grel);
  smpl_pack_atail<<<(M3 * 16) / 256, 256, 0, stream>>>(posedirs, shapedirs, vtemp, Atail);
  dim3 gg((MTILES + 7) / 8, BATCH / 16);
  smpl_gemm<<<gg, 256, 0, stream>>>(posedirs, Atail, Bfrag, vpT);
  dim3 gs((V_CNT + 255) / 256, BATCH);
  smpl_skin<<<gs, 256, 0, stream>>>(W, Grel, vpT, (float*)d_out);
}